// NineNineObserver_71305047048448
// MI455X (gfx1250) — compile-verified
//
#include <hip/hip_runtime.h>
#include <hip/hip_bf16.h>
#include <stdint.h>

// Problem constants (match the reference file).
#define CHANNELS 4096
#define NCOLS    8192
#define KRANK    4915   // int(8192 * 0.6) with python float semantics
#define THREADS  256    // 8 wave32 waves per block

#if defined(__HIP_DEVICE_COMPILE__)
typedef int v4i_t __attribute__((ext_vector_type(4)));
typedef __attribute__((address_space(1))) v4i_t glob_v4i_t;
typedef __attribute__((address_space(3))) v4i_t lds_v4i_t;
#endif

__global__ __launch_bounds__(THREADS)
void nine_select_kernel(const float* __restrict__ x, float* __restrict__ out)
{
    // 32 KB row buffer + 8 wave32-private 256-bucket histograms (8 KB) + scalars.
    __shared__ uint32_t vals[NCOLS];
    __shared__ uint32_t hist[8 * 256];
    __shared__ uint32_t s_minbits;
    __shared__ uint32_t s_prefix;
    __shared__ uint32_t s_krem;

    const int row = blockIdx.x;
    const int tid = threadIdx.x;
    const float* __restrict__ src = x + (size_t)row * NCOLS;

    // ---------------- Stage 1: one-shot copy of the row into LDS ----------------
#if defined(__HIP_DEVICE_COMPILE__) && __has_builtin(__builtin_amdgcn_global_load_async_to_lds_b128)
    {
        // gfx1250 async global->LDS DMA path: each lane moves 16 B per issue.
        // 256 threads * 16 B * 8 issues = 32 KB.
        const char* g = (const char*)src;
        char* l = (char*)vals;
        #pragma unroll
        for (int it = 0; it < (NCOLS * 4) / (THREADS * 16); ++it) {
            const int byteOff = (it * THREADS + tid) * 16;
            __builtin_amdgcn_global_load_async_to_lds_b128(
                (glob_v4i_t*)(uintptr_t)(g + byteOff),
                (lds_v4i_t*)(uint32_t)(uintptr_t)(l + byteOff),
                /*imm offset*/ 0, /*cpol*/ 0);
        }
#if __has_builtin(__builtin_amdgcn_s_wait_asynccnt)
        __builtin_amdgcn_s_wait_asynccnt(0);
#else
        asm volatile("s_wait_asynccnt 0" ::: "memory");
#endif
    }
#else
    {
        // Fallback: vectorized synchronous copy (also what the host pass parses).
        const float4* g4 = (const float4*)src;
        float4* l4 = (float4*)vals;
        for (int i = tid; i < NCOLS / 4; i += THREADS) l4[i] = g4[i];
    }
#endif
    if (tid == 0) { s_minbits = 0x7FFFFFFFu; s_prefix = 0u; s_krem = KRANK; }
    __syncthreads();

    // ---------------- Stage 2: 4-pass radix select on 31 magnitude bits ----------
    // |x| ordering == unsigned ordering of (bits & 0x7FFFFFFF).
    const int      shifts[4] = {23, 15, 7, 0};
    const uint32_t bmasks[4] = {0xFFu, 0xFFu, 0xFFu, 0x7Fu};

    uint32_t prefmask = 0;             // which high bits of the target are known
    uint32_t localMin = 0x7FFFFFFFu;

    for (int p = 0; p < 4; ++p) {
        const int      sh = shifts[p];
        const uint32_t mk = bmasks[p];

        for (int i = tid; i < 8 * 256; i += THREADS) hist[i] = 0;
        __syncthreads();

        const uint32_t pref = s_prefix;
        uint32_t* h = &hist[(tid >> 5) * 256];   // wave32-private histogram

        for (int i = tid; i < NCOLS; i += THREADS) {
            const uint32_t m = vals[i] & 0x7FFFFFFFu;
            if (p == 0) localMin = (m < localMin) ? m : localMin;
            if ((m & prefmask) == pref)
                atomicAdd(&h[(m >> sh) & mk], 1u);
        }
        if (p == 0) atomicMin(&s_minbits, localMin);
        __syncthreads();

        // Reduce the 8 wave-private histograms into hist[0..255].
        {
            uint32_t s = 0;
            #pragma unroll
            for (int w = 0; w < 8; ++w) s += hist[w * 256 + tid];
            hist[tid] = s;
        }
        __syncthreads();

        // Locate the bucket containing the remaining rank.
        if (tid == 0) {
            uint32_t k = s_krem, cum = 0, b = 0;
            for (; b < mk; ++b) {
                const uint32_t c = hist[b];
                if (cum + c > k) break;
                cum += c;
            }
            s_prefix = pref | (b << sh);
            s_krem   = k - cum;
        }
        prefmask |= mk << sh;
        __syncthreads();
    }

    if (tid == 0) {
        out[row]            = __uint_as_float(s_minbits);  // min |x|
        out[CHANNELS + row] = __uint_as_float(s_prefix);   // rank-k |x|
    }
}

extern "C" void kernel_launch(void* const* d_in, const int* in_sizes, int n_in,
                              void* d_out, int out_size, void* d_ws, size_t ws_size,
                              hipStream_t stream)
{
    (void)in_sizes; (void)n_in; (void)out_size; (void)d_ws; (void)ws_size;
    const float* x = (const float*)d_in[0];
    float* out = (float*)d_out;
    nine_select_kernel<<<dim3(CHANNELS), dim3(THREADS), 0, stream>>>(x, out);
}